// BlobbyGyroid_38293928411811
// MI455X (gfx1250) — compile-verified
//
#include <hip/hip_runtime.h>
#include <hip/hip_bf16.h>

typedef __attribute__((ext_vector_type(2))) float v2f;
typedef __attribute__((ext_vector_type(8))) float v8f;

#define JW 4
#define NMB 16
#define KH 8

__device__ __forceinline__ float softplusf_(float x) {
    return fmaxf(x, 0.0f) + log1pf(__expf(-fabsf(x)));
}
__device__ __forceinline__ float sigmoidf_(float x) {
    return 1.0f / (1.0f + __expf(-x));
}

__global__ __launch_bounds__(256)
void blobby_gyroid_kernel(
    const float* __restrict__ p,        const float* __restrict__ t,
    const float* __restrict__ warp_a,   const float* __restrict__ warp_B,
    const float* __restrict__ warp_om,  const float* __restrict__ warp_phi,
    const float* __restrict__ omega,    const float* __restrict__ phi,
    const float* __restrict__ alpha_,   const float* __restrict__ mb_w,
    const float* __restrict__ mb_beta_raw, const float* __restrict__ mb_cbar,
    const float* __restrict__ mb_u,     const float* __restrict__ mb_nu,
    const float* __restrict__ mb_psi,   const float* __restrict__ h_s,
    const float* __restrict__ h_k,      const float* __restrict__ h_w,
    const float* __restrict__ h_zeta,   const float* __restrict__ kappa_,
    const float* __restrict__ bias_b_,  const float* __restrict__ delta_raw_,
    const float* __restrict__ q0,       const float* __restrict__ q1_,
    const float* __restrict__ Q2,       const float* __restrict__ q3_,
    const float* __restrict__ eta_,     const float* __restrict__ w_vec,
    const float* __restrict__ zeta_,    const float* __restrict__ light_dir,
    float* __restrict__ out, int P)
{
    // ---- per-block precompute of derived metaball constants (kills 16
    //      softplus + 3 dot products per point) ----
    __shared__ float sBeta[NMB], sC1[NMB], sC2[NMB], sC3[NMB];
    const int tid = threadIdx.x;
    if (tid < NMB) {
        float cx = mb_cbar[3*tid], cy = mb_cbar[3*tid+1], cz = mb_cbar[3*tid+2];
        float ux = mb_u[3*tid],    uy = mb_u[3*tid+1],    uz = mb_u[3*tid+2];
        sBeta[tid] = softplusf_(mb_beta_raw[tid]) + 0.3f;
        sC1[tid]   = cx*cx + cy*cy + cz*cz;   // |cbar|^2
        sC2[tid]   = cx*ux + cy*uy + cz*uz;   // cbar.u
        sC3[tid]   = ux*ux + uy*uy + uz*uz;   // |u|^2
    }
    __syncthreads();

    const int idx = blockIdx.x * 256 + tid;
    const float px = p[3*idx], py = p[3*idx+1], pz = p[3*idx+2];
    const float tt = t[idx];

    // ---- Stage 1: sinusoidal warp -> q and Jacobian J[c][d] = dq_c/dp_d ----
    float qx = px, qy = py, qz = pz;
    float J00=1.f,J01=0.f,J02=0.f, J10=0.f,J11=1.f,J12=0.f, J20=0.f,J21=0.f,J22=1.f;
    #pragma unroll
    for (int j = 0; j < JW; ++j) {
        const float wo = warp_om[j];
        #pragma unroll
        for (int c = 0; c < 3; ++c) {
            const float b0 = warp_B[j*9 + c*3 + 0];
            const float b1 = warp_B[j*9 + c*3 + 1];
            const float b2 = warp_B[j*9 + c*3 + 2];
            const float a  = warp_a[j*3 + c];
            const float arg = fmaf(b0, px, fmaf(b1, py, fmaf(b2, pz,
                               fmaf(wo, tt, warp_phi[j*3 + c]))));
            float s, co;
            __sincosf(arg, &s, &co);
            const float as = a * s, ac = a * co;
            if (c == 0) { qx += as; J00 = fmaf(ac,b0,J00); J01 = fmaf(ac,b1,J01); J02 = fmaf(ac,b2,J02); }
            else if (c == 1) { qy += as; J10 = fmaf(ac,b0,J10); J11 = fmaf(ac,b1,J11); J12 = fmaf(ac,b2,J12); }
            else { qz += as; J20 = fmaf(ac,b0,J20); J21 = fmaf(ac,b1,J21); J22 = fmaf(ac,b2,J22); }
        }
    }

    // ---- Stage 2: gyroid G and dG/dq ----
    const float wx = omega[0], wy = omega[1], wz = omega[2];
    float su,cu,sv,cv,sw,cw;
    __sincosf(fmaf(wx, qx, phi[0]), &su, &cu);
    __sincosf(fmaf(wy, qy, phi[1]), &sv, &cv);
    __sincosf(fmaf(wz, qz, phi[2]), &sw, &cw);
    const float G   = su*cv + sv*cw + sw*cu - alpha_[0];
    const float gGx = wx * (cu*cv - sw*su);
    const float gGy = wy * (cv*cw - su*sv);
    const float gGz = wz * (cw*cu - sv*sw);

    // ---- Stage 3: harmonics H and dH/dq ----
    float H = 0.f, gHx = 0.f, gHy = 0.f, gHz = 0.f;
    #pragma unroll
    for (int k = 0; k < KH; ++k) {
        const float k0 = h_k[3*k], k1 = h_k[3*k+1], k2 = h_k[3*k+2];
        const float harg = fmaf(k0, qx, fmaf(k1, qy, fmaf(k2, qz,
                            fmaf(h_w[k], tt, h_zeta[k]))));
        float sh, ch;
        __sincosf(harg, &sh, &ch);
        const float hs = h_s[k];
        H = fmaf(hs, sh, H);
        const float f = hs * ch;
        gHx = fmaf(f, k0, gHx); gHy = fmaf(f, k1, gHy); gHz = fmaf(f, k2, gHz);
    }

    // ---- Stage 4: metaballs via WMMA ----------------------------------
    // Gram products q.cbar_n and q.u_n for 32 points x 16 metaballs done as
    // 4x V_WMMA_F32_16X16X4_F32.  A: M=metaball, K=xyz(+pad). B: K=xyz, N=point.
    // D layout gives each lane 8 metaball dots for ONE point; lane halves
    // split metaballs 0-7 / 8-15 and are merged with one shfl_xor(16).
    const int  lane = tid & 31;
    const bool hiH  = lane >= 16;
    const int  m16  = lane & 15;

    v2f Ac, Au, B1, B2;
    Ac.x = hiH ? mb_cbar[3*m16+2] : mb_cbar[3*m16+0];
    Ac.y = hiH ? 0.0f             : mb_cbar[3*m16+1];
    Au.x = hiH ? mb_u[3*m16+2]    : mb_u[3*m16+0];
    Au.y = hiH ? 0.0f             : mb_u[3*m16+1];

    const float qxx = __shfl_xor(qx, 16, 32);
    const float qyx = __shfl_xor(qy, 16, 32);
    const float qzx = __shfl_xor(qz, 16, 32);
    B1.x = hiH ? qzx : qx;  B1.y = hiH ? 0.0f : qy;   // columns = points 0..15
    B2.x = hiH ? qz  : qxx; B2.y = hiH ? 0.0f : qyx;  // columns = points 16..31

    v8f zero8 = {};
    v8f Dc1 = __builtin_amdgcn_wmma_f32_16x16x4_f32(false, Ac, false, B1, (short)0, zero8, false, false);
    v8f Du1 = __builtin_amdgcn_wmma_f32_16x16x4_f32(false, Au, false, B1, (short)0, zero8, false, false);
    v8f Dc2 = __builtin_amdgcn_wmma_f32_16x16x4_f32(false, Ac, false, B2, (short)0, zero8, false, false);
    v8f Du2 = __builtin_amdgcn_wmma_f32_16x16x4_f32(false, Au, false, B2, (short)0, zero8, false, false);

    const float qsq  = qx*qx + qy*qy + qz*qz;
    const float ttx  = __shfl_xor(tt,  16, 32);
    const float qsqx = __shfl_xor(qsq, 16, 32);
    // point A = lane&15  (D*1 columns), point B = (lane&15)+16 (D*2 columns)
    const float tA = hiH ? ttx  : tt,  qsqA = hiH ? qsqx : qsq;
    const float tB = hiH ? tt   : ttx, qsqB = hiH ? qsq  : qsqx;
    const int mbBase = hiH ? 8 : 0;

    float MA=0.f,S0A=0.f,SAx=0.f,SAy=0.f,SAz=0.f;
    float MB=0.f,S0B=0.f,SBx=0.f,SBy=0.f,SBz=0.f;
    #pragma unroll
    for (int r = 0; r < 8; ++r) {
        const int n = mbBase + r;
        const float nu = mb_nu[n], psi = mb_psi[n], w = mb_w[n];
        const float beta = sBeta[n], C1 = sC1[n], C2 = sC2[n], C3 = sC3[n];
        const float cbx = mb_cbar[3*n], cby = mb_cbar[3*n+1], cbz = mb_cbar[3*n+2];
        const float ux  = mb_u[3*n],    uy  = mb_u[3*n+1],    uz  = mb_u[3*n+2];
        {   // element (point A, metaball n)
            const float s  = __sinf(fmaf(nu, tA, psi));
            const float d2 = qsqA - 2.0f*Dc1[r] + C1 + s*(2.0f*(C2 - Du1[r])) + s*s*C3;
            const float e  = __expf(-beta * d2);
            const float we = w*e, wbe = beta*we;
            MA += we; S0A += wbe;
            SAx = fmaf(wbe, fmaf(ux,s,cbx), SAx);
            SAy = fmaf(wbe, fmaf(uy,s,cby), SAy);
            SAz = fmaf(wbe, fmaf(uz,s,cbz), SAz);
        }
        {   // element (point B, metaball n)
            const float s  = __sinf(fmaf(nu, tB, psi));
            const float d2 = qsqB - 2.0f*Dc2[r] + C1 + s*(2.0f*(C2 - Du2[r])) + s*s*C3;
            const float e  = __expf(-beta * d2);
            const float we = w*e, wbe = beta*we;
            MB += we; S0B += wbe;
            SBx = fmaf(wbe, fmaf(ux,s,cbx), SBx);
            SBy = fmaf(wbe, fmaf(uy,s,cby), SBy);
            SBz = fmaf(wbe, fmaf(uz,s,cbz), SBz);
        }
    }
    // merge metaball halves: own point's other-half partial lives in lane^16
    const float ownM  = hiH ? MB  : MA,  othM  = hiH ? MA  : MB;
    const float ownS0 = hiH ? S0B : S0A, othS0 = hiH ? S0A : S0B;
    const float ownSx = hiH ? SBx : SAx, othSx = hiH ? SAx : SBx;
    const float ownSy = hiH ? SBy : SAy, othSy = hiH ? SAy : SBy;
    const float ownSz = hiH ? SBz : SAz, othSz = hiH ? SAz : SBz;
    const float M  = ownM  + __shfl_xor(othM,  16, 32);
    const float S0 = ownS0 + __shfl_xor(othS0, 16, 32);
    const float Sx = ownSx + __shfl_xor(othSx, 16, 32);
    const float Sy = ownSy + __shfl_xor(othSy, 16, 32);
    const float Sz = ownSz + __shfl_xor(othSz, 16, 32);

    const float gMx = -2.0f * (qx*S0 - Sx);
    const float gMy = -2.0f * (qy*S0 - Sy);
    const float gMz = -2.0f * (qz*S0 - Sz);

    // ---- Stage 5: assemble F, grad, outputs ----
    const float kap  = kappa_[0];
    const float F    = G + kap * log1pf(M) + H - bias_b_[0];
    const float gfac = kap / (1.0f + M);
    const float dFx  = gGx + gfac*gMx + gHx;
    const float dFy  = gGy + gfac*gMy + gHy;
    const float dFz  = gGz + gfac*gMz + gHz;
    // grad_p[d] = sum_c J[c][d] * dF/dq_c
    const float gpx = J00*dFx + J10*dFy + J20*dFz;
    const float gpy = J01*dFx + J11*dFy + J21*dFz;
    const float gpz = J02*dFx + J12*dFy + J22*dFz;
    const float nrm = sqrtf(gpx*gpx + gpy*gpy + gpz*gpz);
    const float inv = 1.0f / fmaxf(nrm, 1e-12f);
    const float nx = gpx*inv, ny = gpy*inv, nz = gpz*inv;

    const float delta = softplusf_(delta_raw_[0]) + 0.001f;
    const float sigma = softplusf_(-F / delta);
    const float ndotl = nx*light_dir[0] + ny*light_dir[1] + nz*light_dir[2];
    const float emis  = __sinf(fmaf(eta_[0],
                          qx*w_vec[0] + qy*w_vec[1] + qz*w_vec[2],
                          zeta_[0] * tt));
    const float q1 = q1_[0], q3 = q3_[0];
    const float r0 = sigmoidf_(q0[0] + q1*ndotl + Q2[0]*nx + Q2[1]*ny + Q2[2]*nz + q3*emis);
    const float r1 = sigmoidf_(q0[1] + q1*ndotl + Q2[3]*nx + Q2[4]*ny + Q2[5]*nz + q3*emis);
    const float r2 = sigmoidf_(q0[2] + q1*ndotl + Q2[6]*nx + Q2[7]*ny + Q2[8]*nz + q3*emis);

    // outputs: sigma[P] | rgb[P,3] | F[P] | n[P,3]
    out[idx] = sigma;
    float* rgb = out + (size_t)P;
    rgb[3*idx+0] = r0; rgb[3*idx+1] = r1; rgb[3*idx+2] = r2;
    out[(size_t)4*P + idx] = F;
    float* no = out + (size_t)5*P;
    no[3*idx+0] = nx; no[3*idx+1] = ny; no[3*idx+2] = nz;
}

extern "C" void kernel_launch(void* const* d_in, const int* in_sizes, int n_in,
                              void* d_out, int out_size, void* d_ws, size_t ws_size,
                              hipStream_t stream) {
    const int P = in_sizes[0] / 3;   // P = 2^21, divisible by 256 -> guard-free
    dim3 block(256), grid(P / 256);
    blobby_gyroid_kernel<<<grid, block, 0, stream>>>(
        (const float*)d_in[0],  (const float*)d_in[1],  (const float*)d_in[2],
        (const float*)d_in[3],  (const float*)d_in[4],  (const float*)d_in[5],
        (const float*)d_in[6],  (const float*)d_in[7],  (const float*)d_in[8],
        (const float*)d_in[9],  (const float*)d_in[10], (const float*)d_in[11],
        (const float*)d_in[12], (const float*)d_in[13], (const float*)d_in[14],
        (const float*)d_in[15], (const float*)d_in[16], (const float*)d_in[17],
        (const float*)d_in[18], (const float*)d_in[19], (const float*)d_in[20],
        (const float*)d_in[21], (const float*)d_in[22], (const float*)d_in[23],
        (const float*)d_in[24], (const float*)d_in[25], (const float*)d_in[26],
        (const float*)d_in[27], (const float*)d_in[28], (const float*)d_in[29],
        (float*)d_out, P);
}